// PolypharmacyHGT_24180665876644
// MI455X (gfx1250) — compile-verified
//
#include <hip/hip_runtime.h>
#include <hip/hip_bf16.h>
#include <math.h>

typedef float v2f __attribute__((ext_vector_type(2)));
typedef float v8f __attribute__((ext_vector_type(8)));

#define DMODEL 128
#define NHEAD  8
#define DHEAD  16
#define N_DRUG 4000
#define N_PROT 20000
#define N_SE   964
#define E1     100000
#define E3     300000
#define E4     150000
#define M_PW   300000
#define G_PW   50000
#define N_PAIR 150000

#define ENC_NEG_INF 0x007FFFFFu   // order-encoding of -inf

// ---------------- device helpers ----------------
// tanh-approx GELU, branch-free: 0.5x(1+tanh(u)) == x*sigmoid(2u)
__device__ __forceinline__ float gelu_f(float x) {
    const float c2 = 1.5957691216057308f;   // 2*sqrt(2/pi)
    float u2 = c2 * (x + 0.044715f * x * x * x);
    return x / (1.0f + __expf(-u2));
}
__device__ __forceinline__ float sigmoid_f(float x) {
    return 1.0f / (1.0f + __expf(-x));
}
__device__ __forceinline__ unsigned enc_f(float f) {
    unsigned u = __float_as_uint(f);
    return (u & 0x80000000u) ? ~u : (u | 0x80000000u);
}
__device__ __forceinline__ float dec_f(unsigned u) {
    u = (u & 0x80000000u) ? (u & 0x7fffffffu) : ~u;
    return __uint_as_float(u);
}
__device__ __forceinline__ float wave_sum32(float v) {
    #pragma unroll
    for (int m = 16; m >= 1; m >>= 1) v += __shfl_xor(v, m);
    return v;
}

// ---------------- WMMA GEMM: Y[M][128] = act(pre(X)[M][K] @ W[K][128] + b) ----------------
// Compile-time ACT (0 none / 1 relu), INGELU (gelu on X load), TRANSB (B[k][n] = W[n*K+k]).
// Each wave owns a 32(M)x16(N) tile: one B fragment feeds two independent WMMAs.
template<int ACT, int INGELU, int TRANSB>
__global__ void gemm128_t(const float* __restrict__ X, int ldx,
                          const float* __restrict__ W,
                          const float* __restrict__ bias,
                          float* __restrict__ Y, int M, int K)
{
    int wave = threadIdx.x >> 5;
    int lane = threadIdx.x & 31;
    int half = lane >> 4, l15 = lane & 15;
    int row0 = blockIdx.x * 32;
    int col0 = wave * 16;
    if (row0 >= M) return;
    int ar0 = row0 + l15;      if (ar0 >= M) ar0 = M - 1;
    int ar1 = row0 + 16 + l15; if (ar1 >= M) ar1 = M - 1;
    const float* Xr0 = X + (size_t)ar0 * ldx;
    const float* Xr1 = X + (size_t)ar1 * ldx;
    int col = col0 + l15;
    v8f c0 = {0.f,0.f,0.f,0.f,0.f,0.f,0.f,0.f};
    v8f c1 = {0.f,0.f,0.f,0.f,0.f,0.f,0.f,0.f};
    for (int k = 0; k < K; k += 4) {
        int kk = k + half * 2;
        v2f b;
        if (TRANSB) { b.x = W[(size_t)col * K + kk];      b.y = W[(size_t)col * K + kk + 1]; }
        else        { b.x = W[(size_t)kk * DMODEL + col]; b.y = W[(size_t)(kk + 1) * DMODEL + col]; }
        v2f a0, a1;
        a0.x = Xr0[kk]; a0.y = Xr0[kk + 1];
        a1.x = Xr1[kk]; a1.y = Xr1[kk + 1];
        if (INGELU) {
            a0.x = gelu_f(a0.x); a0.y = gelu_f(a0.y);
            a1.x = gelu_f(a1.x); a1.y = gelu_f(a1.y);
        }
        c0 = __builtin_amdgcn_wmma_f32_16x16x4_f32(false, a0, false, b, (short)0, c0, false, false);
        c1 = __builtin_amdgcn_wmma_f32_16x16x4_f32(false, a1, false, b, (short)0, c1, false, false);
    }
    float bv = bias ? bias[col] : 0.0f;
    #pragma unroll
    for (int r = 0; r < 8; ++r) {
        int row = row0 + r + half * 8;
        float v0 = c0[r] + bv;
        float v1 = c1[r] + bv;
        if (ACT == 1) { v0 = fmaxf(v0, 0.0f); v1 = fmaxf(v1, 0.0f); }
        if (row < M)      Y[(size_t)row * DMODEL + col]        = v0;
        if (row + 16 < M) Y[(size_t)(row + 16) * DMODEL + col] = v1;
    }
}

static inline void launch_gemm(const float* X, int ldx, const float* W, const float* bias,
                               float* Y, int M, int K, int act, int ig, int tb, hipStream_t s)
{
    dim3 blk(256), grd((M + 31) / 32);
    if (ig)       gemm128_t<0,1,0><<<grd, blk, 0, s>>>(X, ldx, W, bias, Y, M, K);
    else if (tb)  gemm128_t<0,0,1><<<grd, blk, 0, s>>>(X, ldx, W, bias, Y, M, K);
    else if (act) gemm128_t<1,0,0><<<grd, blk, 0, s>>>(X, ldx, W, bias, Y, M, K);
    else          gemm128_t<0,0,0><<<grd, blk, 0, s>>>(X, ldx, W, bias, Y, M, K);
}

// ---------------- HGT edge logits ----------------
// logit[e,h] = ((K[s]·a[h]) . Q[d]) * p[h]/4.  One wave = 16 edges x 8 heads.
// Transposed WMMA: D = a[h]^T x Kgathered^T  ->  krel^T with f in regs, edge in lanes.
// The f-reduction is then 8 in-lane FMAs vs two float4 loads of Q, plus ONE xor-16 shuffle.
__global__ void edge_logits_kernel(const float* __restrict__ Ksrc,
                                   const float* __restrict__ Qdst,
                                   const int* __restrict__ src,
                                   const int* __restrict__ dst,
                                   const float* __restrict__ rel_a,  // [8][16][16]
                                   const float* __restrict__ rel_p,  // [8]
                                   float* __restrict__ logits,       // [E][8]
                                   int E)
{
    __shared__ float sA[NHEAD * 256];
    for (int i = threadIdx.x; i < NHEAD * 256; i += 256) sA[i] = rel_a[i];
    __syncthreads();

    int wave = threadIdx.x >> 5, lane = threadIdx.x & 31;
    int half = lane >> 4, l15 = lane & 15;
    int e0 = (blockIdx.x * 8 + wave) * 16;
    if (e0 >= E) return;
    int eL = e0 + l15; if (eL >= E) eL = E - 1;
    const float* Krow = Ksrc + (size_t)src[eL] * DMODEL;   // edge l15's K row
    const float* Qrow = Qdst + (size_t)dst[eL] * DMODEL;   // edge l15's Q row
    const float scale = 0.25f; // 1/sqrt(16)
    for (int h = 0; h < NHEAD; ++h) {
        const float* Ah = sA + h * 256;
        v8f c = {0.f,0.f,0.f,0.f,0.f,0.f,0.f,0.f};
        #pragma unroll
        for (int s = 0; s < 4; ++s) {
            int kk = s * 4 + half * 2;
            v2f a, b;
            // A = a[h]^T : A[f=l15][d=kk] = a[d][f]
            a.x = Ah[kk * DHEAD + l15];  a.y = Ah[(kk + 1) * DHEAD + l15];
            // B = K^T    : B[d=kk][e=l15] = K[e][h*16+d]
            b.x = Krow[h * DHEAD + kk];  b.y = Krow[h * DHEAD + kk + 1];
            c = __builtin_amdgcn_wmma_f32_16x16x4_f32(false, a, false, b, (short)0, c, false, false);
        }
        // c[r] = krel[e=l15][f = r + 8*half]
        float ph = rel_p[h] * scale;
        float4 q0 = *(const float4*)(Qrow + h * DHEAD + half * 8);
        float4 q1 = *(const float4*)(Qrow + h * DHEAD + half * 8 + 4);
        float prt = c[0] * q0.x + c[1] * q0.y + c[2] * q0.z + c[3] * q0.w
                  + c[4] * q1.x + c[5] * q1.y + c[6] * q1.z + c[7] * q1.w;
        prt += __shfl_xor(prt, 16);          // combine f-halves
        if (half == 0 && (e0 + l15) < E)
            logits[(size_t)(e0 + l15) * NHEAD + h] = prt * ph;
    }
}

// ---------------- segmented softmax helpers (compile-time H: 8 for HGT, 1 for pooling) ----------------
template<int H>
__global__ void seg_max_t(const float* __restrict__ logit, const int* __restrict__ dst,
                          unsigned* __restrict__ mx, long n)
{
    long i = (long)blockIdx.x * 256 + threadIdx.x;
    if (i >= n) return;
    long e = i / H; int h = (int)(i - e * H);
    atomicMax(&mx[(size_t)dst[e] * H + h], enc_f(logit[i]));
}

template<int H>
__global__ void seg_den_t(float* __restrict__ logit, const int* __restrict__ dst,
                          const unsigned* __restrict__ mx, float* __restrict__ den, long n)
{
    long i = (long)blockIdx.x * 256 + threadIdx.x;
    if (i >= n) return;
    long e = i / H; int h = (int)(i - e * H);
    unsigned mu = mx[(size_t)dst[e] * H + h];
    float m = (mu == ENC_NEG_INF) ? 0.0f : dec_f(mu);
    float w = __expf(logit[i] - m);
    logit[i] = w;                                  // overwrite logits with exp-weights
    atomicAdd(&den[(size_t)dst[e] * H + h], w);
}

// ---------------- HGT edge aggregation: agg[d] += alpha * (V[s]·m[h]) ----------------
// Untransposed orientation on purpose: f in lanes => 16 consecutive-float atomics per edge.
__global__ void edge_agg_kernel(const float* __restrict__ Vsrc,
                                const int* __restrict__ src,
                                const int* __restrict__ dst,
                                const float* __restrict__ rel_m,  // [8][16][16]
                                const float* __restrict__ w,      // [E][8] exp-weights
                                const float* __restrict__ den,    // [Ndst][8]
                                float* __restrict__ agg,          // [Ndst][128]
                                int E)
{
    __shared__ float sM[NHEAD * 256];
    for (int i = threadIdx.x; i < NHEAD * 256; i += 256) sM[i] = rel_m[i];
    __syncthreads();

    int wave = threadIdx.x >> 5, lane = threadIdx.x & 31;
    int half = lane >> 4, l15 = lane & 15;
    int e0 = (blockIdx.x * 8 + wave) * 16;
    if (e0 >= E) return;
    int eA = e0 + l15; if (eA >= E) eA = E - 1;
    const float* Vrow = Vsrc + (size_t)src[eA] * DMODEL;
    for (int h = 0; h < NHEAD; ++h) {
        const float* Mh = sM + h * 256;
        v8f c = {0.f,0.f,0.f,0.f,0.f,0.f,0.f,0.f};
        #pragma unroll
        for (int s = 0; s < 4; ++s) {
            int kk = s * 4 + half * 2;
            v2f a, b;
            a.x = Vrow[h * DHEAD + kk];  a.y = Vrow[h * DHEAD + kk + 1];
            b.x = Mh[kk * DHEAD + l15];  b.y = Mh[(kk + 1) * DHEAD + l15];
            c = __builtin_amdgcn_wmma_f32_16x16x4_f32(false, a, false, b, (short)0, c, false, false);
        }
        #pragma unroll
        for (int r = 0; r < 8; ++r) {
            int er = e0 + r + half * 8;
            if (er >= E) continue;
            int dn = dst[er];
            float alpha = w[(size_t)er * NHEAD + h] / fmaxf(den[(size_t)dn * NHEAD + h], 1e-9f);
            atomicAdd(&agg[(size_t)dn * DMODEL + h * DHEAD + l15], c[r] * alpha);
        }
    }
}

// ---------------- combine + LayerNorm (wave per row); o==null -> pure LN ----------------
__global__ void ln_kernel(const float* __restrict__ x, const float* __restrict__ o,
                          const float* __restrict__ skip,
                          const float* __restrict__ g, const float* __restrict__ bb,
                          float* __restrict__ out, int N)
{
    int wave = threadIdx.x >> 5, lane = threadIdx.x & 31;
    int row = blockIdx.x * 8 + wave;
    if (row >= N) return;
    float4 xv = *(const float4*)(x + (size_t)row * DMODEL + lane * 4);
    float t0 = xv.x, t1 = xv.y, t2 = xv.z, t3 = xv.w;
    if (o) {
        float gate = sigmoid_f(skip[0]);
        float4 ov = *(const float4*)(o + (size_t)row * DMODEL + lane * 4);
        float gh = 2.0f - gate;   // gate*o + (1-gate)*h + h
        t0 = gate * ov.x + gh * t0;
        t1 = gate * ov.y + gh * t1;
        t2 = gate * ov.z + gh * t2;
        t3 = gate * ov.w + gh * t3;
    }
    float mean = wave_sum32(t0 + t1 + t2 + t3) * (1.0f / 128.0f);
    float d0 = t0 - mean, d1 = t1 - mean, d2 = t2 - mean, d3 = t3 - mean;
    float var = wave_sum32(d0 * d0 + d1 * d1 + d2 * d2 + d3 * d3) * (1.0f / 128.0f);
    float rstd = rsqrtf(var + 1e-5f);
    int cidx = lane * 4;
    float4 r;
    r.x = d0 * rstd * g[cidx + 0] + bb[cidx + 0];
    r.y = d1 * rstd * g[cidx + 1] + bb[cidx + 1];
    r.z = d2 * rstd * g[cidx + 2] + bb[cidx + 2];
    r.w = d3 * rstd * g[cidx + 3] + bb[cidx + 3];
    *(float4*)(out + (size_t)row * DMODEL + lane * 4) = r;
}

// ---------------- pathway pooling ----------------
__global__ void pool_score_kernel(const float* __restrict__ hp, const int* __restrict__ pw_prot,
                                  const float* __restrict__ aw, float* __restrict__ sc, int M)
{
    int wave = threadIdx.x >> 5, lane = threadIdx.x & 31;
    int m = blockIdx.x * 8 + wave;
    if (m >= M) return;
    float4 hv = *((const float4*)(hp + (size_t)pw_prot[m] * DMODEL) + lane);
    float4 av = *((const float4*)aw + lane);
    float s = wave_sum32(hv.x * av.x + hv.y * av.y + hv.z * av.z + hv.w * av.w);
    if (lane == 0) sc[m] = s;
}

__global__ void pool_agg_kernel(const float* __restrict__ hp, const int* __restrict__ pw_prot,
                                const int* __restrict__ pw_group, const float* __restrict__ w,
                                const float* __restrict__ den, float* __restrict__ pooled, int M)
{
    int wave = threadIdx.x >> 5, lane = threadIdx.x & 31;
    int m = blockIdx.x * 8 + wave;
    if (m >= M) return;
    int g = pw_group[m];
    float alpha = w[m] / fmaxf(den[g], 1e-9f);
    float4 hv = *((const float4*)(hp + (size_t)pw_prot[m] * DMODEL) + lane);
    float* pp = pooled + (size_t)g * DMODEL + lane * 4;
    atomicAdd(pp + 0, alpha * hv.x);
    atomicAdd(pp + 1, alpha * hv.y);
    atomicAdd(pp + 2, alpha * hv.z);
    atomicAdd(pp + 3, alpha * hv.w);
}

__global__ void group2drug_kernel(const float* __restrict__ pooled, const int* __restrict__ gd,
                                  float* __restrict__ sums, float* __restrict__ cnt, int G)
{
    int wave = threadIdx.x >> 5, lane = threadIdx.x & 31;
    int g = blockIdx.x * 8 + wave;
    if (g >= G) return;
    int dr = gd[g];
    float4 pv = *((const float4*)(pooled + (size_t)g * DMODEL) + lane);
    float* s = sums + (size_t)dr * DMODEL + lane * 4;
    atomicAdd(s + 0, pv.x);
    atomicAdd(s + 1, pv.y);
    atomicAdd(s + 2, pv.z);
    atomicAdd(s + 3, pv.w);
    if (lane == 0) atomicAdd(&cnt[dr], 1.0f);
}

__global__ void fp_div_kernel(const float* __restrict__ sums, const float* __restrict__ cnt,
                              float* __restrict__ fp, long n)
{
    long i = (long)blockIdx.x * 256 + threadIdx.x;
    if (i >= n) return;
    long row = i >> 7;
    fp[i] = sums[i] / fmaxf(cnt[row], 1.0f);
}

__global__ void concat_kernel(const float* __restrict__ hd, const float* __restrict__ fp,
                              float* __restrict__ zcat, long n)
{
    long i = (long)blockIdx.x * 256 + threadIdx.x;
    if (i >= n) return;
    long row = i >> 8; int c = (int)(i & 255);
    zcat[i] = (c < 128) ? hd[row * 128 + c] : fp[row * 128 + (c - 128)];
}

// ---------------- DEDICOM decode: sigmoid(sum zi * dg^2 * (z R^T)[j]) ----------------
__global__ void decode_kernel(const float* __restrict__ z, const float* __restrict__ ZR,
                              const float* __restrict__ Dg,
                              const int* __restrict__ pi, const int* __restrict__ pj,
                              const int* __restrict__ se, float* __restrict__ out, int NP)
{
    int wave = threadIdx.x >> 5, lane = threadIdx.x & 31;
    int p = blockIdx.x * 8 + wave;
    if (p >= NP) return;
    float4 a = *((const float4*)(z  + (size_t)pi[p] * DMODEL) + lane);
    float4 b = *((const float4*)(ZR + (size_t)pj[p] * DMODEL) + lane);
    float4 d = *((const float4*)(Dg + (size_t)se[p] * DMODEL) + lane);
    float acc = a.x * b.x * d.x * d.x + a.y * b.y * d.y * d.y
              + a.z * b.z * d.z * d.z + a.w * b.w * d.w * d.w;
    acc = wave_sum32(acc);
    if (lane == 0) out[p] = sigmoid_f(acc);
}

__global__ void fill_kernel(unsigned* __restrict__ p, unsigned v, long n)
{
    long i = (long)blockIdx.x * 256 + threadIdx.x;
    if (i < n) p[i] = v;
}

// ---------------- param leaf layout (jax pytree order: sorted dict keys) ----------------
static const long LEAF_SIZES[49] = {
    964L*128, 128L*128,                               // dec.Dg, dec.R
    256L*128, 128L*128, 128, 128, 128, 128,           // fuse: W1,W2,b1,b2,ln_b,ln_g
    2L*128*128,2L*128*128,2L*128*128,2L*128*128,      // hgt.drug: Wa,Wk,Wq,Wv
    2L*128,2L*128,2L*128,2L*128,2L*128,2L*128,2,      //           ba,bk,bq,bv,ln_b,ln_g,skip
    2L*128*128,2L*128*128,2L*128*128,2L*128*128,      // hgt.protein
    2L*128,2L*128,2L*128,2L*128,2L*128,2L*128,2,
    128L*128, 128, 128,                               // pool: W, attn_w, b
    512L*128, 128, 256L*128, 128,                     // proj: drug.W, drug.b, prot.W, prot.b
    2L*8*16*16, 2L*8*16*16, 2L*8,                     // rel r1: a,m,p
    2L*8*16*16, 2L*8*16*16, 2L*8,                     // r2
    2L*8*16*16, 2L*8*16*16, 2L*8,                     // r3
    2L*8*16*16, 2L*8*16*16, 2L*8                      // r4
};

extern "C" void kernel_launch(void* const* d_in, const int* in_sizes, int n_in,
                              void* d_out, int out_size, void* d_ws, size_t ws_size,
                              hipStream_t stream)
{
    (void)in_sizes; (void)out_size; (void)ws_size;
    // ---- resolve parameter leaves (either one leaf per d_in slot, or one flat buffer) ----
    const float* leaf[49];
    int base;
    if (n_in >= 49 + 16) {
        for (int i = 0; i < 49; ++i) leaf[i] = (const float*)d_in[i];
        base = 49;
    } else {
        const float* p0 = (const float*)d_in[0];
        long off = 0;
        for (int i = 0; i < 49; ++i) { leaf[i] = p0 + off; off += LEAF_SIZES[i]; }
        base = 1;
    }
    const float* x_drug = (const float*)d_in[base + 0];
    const float* x_prot = (const float*)d_in[base + 1];
    const int* e_src[4] = { (const int*)d_in[base + 2], (const int*)d_in[base + 4],
                            (const int*)d_in[base + 6], (const int*)d_in[base + 8] };
    const int* e_dst[4] = { (const int*)d_in[base + 3], (const int*)d_in[base + 5],
                            (const int*)d_in[base + 7], (const int*)d_in[base + 9] };
    const int* pw_prot       = (const int*)d_in[base + 10];
    const int* pw_group      = (const int*)d_in[base + 11];
    const int* pw_group_drug = (const int*)d_in[base + 12];
    const int* pair_i        = (const int*)d_in[base + 13];
    const int* pair_j        = (const int*)d_in[base + 14];
    const int* se_idx        = (const int*)d_in[base + 15];
    float* out = (float*)d_out;

    // ---- workspace layout (floats) ----
    float* W = (float*)d_ws;
    long off = 0;
    #define WSALLOC(name, n) float* name = W + off; off += (n)
    WSALLOC(h_drug, 4000L*128);   WSALLOC(h_prot, 20000L*128);
    WSALLOC(Kd, 4000L*128);  WSALLOC(Qd, 4000L*128);  WSALLOC(Vd, 4000L*128);
    WSALLOC(Kp, 20000L*128); WSALLOC(Qp, 20000L*128); WSALLOC(Vp, 20000L*128);
    WSALLOC(agg_d, 4000L*128);    WSALLOC(agg_p, 20000L*128);
    WSALLOC(o_d, 4000L*128);      WSALLOC(o_p, 20000L*128);
    WSALLOC(logits, 650000L*8);
    WSALLOC(mx_d_f, 4000L*8);     WSALLOC(den_d, 4000L*8);
    WSALLOC(mx_p_f, 20000L*8);    WSALLOC(den_p, 20000L*8);
    WSALLOC(sc, 300000L);
    WSALLOC(mxg_f, 50000L);       WSALLOC(deng, 50000L);
    WSALLOC(pooled, 50000L*128);
    WSALLOC(sums, 4000L*128);     WSALLOC(cnt, 4000L);
    WSALLOC(fp, 4000L*128);
    WSALLOC(zcat, 4000L*256);
    WSALLOC(z1, 4000L*128);       WSALLOC(z, 4000L*128);   WSALLOC(ZR, 4000L*128);
    #undef WSALLOC
    unsigned* mx_d = (unsigned*)mx_d_f;
    unsigned* mx_p = (unsigned*)mx_p_f;
    unsigned* mxg  = (unsigned*)mxg_f;

    dim3 blk(256);
    #define FILL(p, v, n) \
        fill_kernel<<<dim3((unsigned)(((n)+255)/256)), blk, 0, stream>>>((unsigned*)(p), v, (long)(n))

    // ---- input projections (ReLU) ----
    launch_gemm(x_drug, 512, leaf[33], leaf[34], h_drug, N_DRUG, 512, 1, 0, 0, stream);
    launch_gemm(x_prot, 256, leaf[35], leaf[36], h_prot, N_PROT, 256, 1, 0, 0, stream);

    // relation tables: r1 drug->prot, r2 prot->drug, r3 prot->prot, r4 drug->drug
    const int   relE[4]     = { E1, E1, E3, E4 };
    const long  relLgOff[4] = { 0L, 100000L*8, 200000L*8, 500000L*8 };
    const int   relAleaf[4] = { 37, 40, 43, 46 };

    for (int l = 0; l < 2; ++l) {
        long wOff = (long)l * 128 * 128, bOff = (long)l * 128;
        // K/Q/V projections
        launch_gemm(h_drug, 128, leaf[9]  + wOff, leaf[13] + bOff, Kd, N_DRUG, 128, 0, 0, 0, stream);
        launch_gemm(h_drug, 128, leaf[10] + wOff, leaf[14] + bOff, Qd, N_DRUG, 128, 0, 0, 0, stream);
        launch_gemm(h_drug, 128, leaf[11] + wOff, leaf[15] + bOff, Vd, N_DRUG, 128, 0, 0, 0, stream);
        launch_gemm(h_prot, 128, leaf[20] + wOff, leaf[24] + bOff, Kp, N_PROT, 128, 0, 0, 0, stream);
        launch_gemm(h_prot, 128, leaf[21] + wOff, leaf[25] + bOff, Qp, N_PROT, 128, 0, 0, 0, stream);
        launch_gemm(h_prot, 128, leaf[22] + wOff, leaf[26] + bOff, Vp, N_PROT, 128, 0, 0, 0, stream);
        // reset softmax/agg state
        FILL(mx_d, ENC_NEG_INF, 4000L*8);  FILL(den_d, 0u, 4000L*8);
        FILL(mx_p, ENC_NEG_INF, 20000L*8); FILL(den_p, 0u, 20000L*8);
        FILL(agg_d, 0u, 4000L*128);        FILL(agg_p, 0u, 20000L*128);

        for (int r = 0; r < 4; ++r) {
            const float* Ks = (r == 0 || r == 3) ? Kd : Kp;
            const float* Qx = (r == 0 || r == 2) ? Qp : Qd;
            const float* ra = leaf[relAleaf[r]]     + (long)l * 2048;
            const float* rp = leaf[relAleaf[r] + 2] + (long)l * 8;
            edge_logits_kernel<<<dim3((relE[r] + 127) / 128), blk, 0, stream>>>(
                Ks, Qx, e_src[r], e_dst[r], ra, rp, logits + relLgOff[r], relE[r]);
        }
        for (int r = 0; r < 4; ++r) {
            unsigned* mx = (r == 0 || r == 2) ? mx_p : mx_d;
            long n = (long)relE[r] * 8;
            seg_max_t<8><<<dim3((unsigned)((n + 255) / 256)), blk, 0, stream>>>(
                logits + relLgOff[r], e_dst[r], mx, n);
        }
        for (int r = 0; r < 4; ++r) {
            unsigned* mx = (r == 0 || r == 2) ? mx_p : mx_d;
            float* den   = (r == 0 || r == 2) ? den_p : den_d;
            long n = (long)relE[r] * 8;
            seg_den_t<8><<<dim3((unsigned)((n + 255) / 256)), blk, 0, stream>>>(
                logits + relLgOff[r], e_dst[r], mx, den, n);
        }
        for (int r = 0; r < 4; ++r) {
            const float* Vs = (r == 0 || r == 3) ? Vd : Vp;
            float* den = (r == 0 || r == 2) ? den_p : den_d;
            float* agg = (r == 0 || r == 2) ? agg_p : agg_d;
            const float* rm = leaf[relAleaf[r] + 1] + (long)l * 2048;
            edge_agg_kernel<<<dim3((relE[r] + 127) / 128), blk, 0, stream>>>(
                Vs, e_src[r], e_dst[r], rm, logits + relLgOff[r], den, agg, relE[r]);
        }
        // o = gelu(agg) @ Wa + ba ; then gated skip + residual + LN
        launch_gemm(agg_d, 128, leaf[8]  + wOff, leaf[12] + bOff, o_d, N_DRUG, 128, 0, 1, 0, stream);
        launch_gemm(agg_p, 128, leaf[19] + wOff, leaf[23] + bOff, o_p, N_PROT, 128, 0, 1, 0, stream);
        ln_kernel<<<dim3((N_DRUG + 7) / 8), blk, 0, stream>>>(
            h_drug, o_d, leaf[18] + l, leaf[17] + bOff, leaf[16] + bOff, h_drug, N_DRUG);
        ln_kernel<<<dim3((N_PROT + 7) / 8), blk, 0, stream>>>(
            h_prot, o_p, leaf[29] + l, leaf[28] + bOff, leaf[27] + bOff, h_prot, N_PROT);
    }

    // ---- pathway attention pooling ----
    FILL(mxg, ENC_NEG_INF, G_PW); FILL(deng, 0u, G_PW);
    FILL(pooled, 0u, (long)G_PW * 128);
    FILL(sums, 0u, 4000L * 128);  FILL(cnt, 0u, 4000L);
    pool_score_kernel<<<dim3((M_PW + 7) / 8), blk, 0, stream>>>(h_prot, pw_prot, leaf[31], sc, M_PW);
    seg_max_t<1><<<dim3((M_PW + 255) / 256), blk, 0, stream>>>(sc, pw_group, mxg, M_PW);
    seg_den_t<1><<<dim3((M_PW + 255) / 256), blk, 0, stream>>>(sc, pw_group, mxg, deng, M_PW);
    pool_agg_kernel<<<dim3((M_PW + 7) / 8), blk, 0, stream>>>(h_prot, pw_prot, pw_group, sc, deng, pooled, M_PW);
    group2drug_kernel<<<dim3((G_PW + 7) / 8), blk, 0, stream>>>(pooled, pw_group_drug, sums, cnt, G_PW);
    fp_div_kernel<<<dim3((4000 * 128 + 255) / 256), blk, 0, stream>>>(sums, cnt, fp, 4000L * 128);
    launch_gemm(fp, 128, leaf[30], leaf[32], sums, N_DRUG, 128, 1, 0, 0, stream);   // fp_final -> sums (reuse)

    // ---- fusion MLP + LN ----
    concat_kernel<<<dim3((4000 * 256 + 255) / 256), blk, 0, stream>>>(h_drug, sums, zcat, 4000L * 256);
    launch_gemm(zcat, 256, leaf[2], leaf[4], z1, N_DRUG, 256, 1, 0, 0, stream);
    launch_gemm(z1, 128, leaf[3], leaf[5], z, N_DRUG, 128, 0, 0, 0, stream);
    ln_kernel<<<dim3((N_DRUG + 7) / 8), blk, 0, stream>>>(
        z, (const float*)nullptr, (const float*)nullptr, leaf[7], leaf[6], z, N_DRUG);

    // ---- DEDICOM decode: ZR = z @ R^T once for 4000 drugs, then wave-per-pair ----
    launch_gemm(z, 128, leaf[1], (const float*)nullptr, ZR, N_DRUG, 128, 0, 0, 1, stream);
    decode_kernel<<<dim3((N_PAIR + 7) / 8), blk, 0, stream>>>(
        z, ZR, leaf[0], pair_i, pair_j, se_idx, out, N_PAIR);

    #undef FILL
}